// ChainCRF4NestedNER_86053964743154
// MI455X (gfx1250) — compile-verified
//
#include <hip/hip_runtime.h>
#include <hip/hip_bf16.h>

typedef __attribute__((ext_vector_type(2))) float v2f;
typedef __attribute__((ext_vector_type(8))) float v8f;

#define NL        31
#define LPAD      32
#define IDX_O     28
#define IDX_EOS   29
#define IDX_BOS   30
#define NEGV      (-10000.0f)
#define NINF      (-1e30f)
#define B_DIM     64
#define T_DIM     512
#define D_DIM     768

// ---------------------------------------------------------------------------
// Kernel 0: build zero-padded weights wpad[32][768] (row 31 = 0) so the GEMM
// K-loop has no divergent guard for the pad column.
// ---------------------------------------------------------------------------
__global__ __launch_bounds__(256) void crf_prep_weights(
    const float* __restrict__ state_w,  // (31, 768)
    float* __restrict__ wpad)           // (32, 768)
{
    const int idx = blockIdx.x * 256 + threadIdx.x;   // 0 .. 32*768-1
    const int c = idx / D_DIM;
    wpad[idx] = (c < NL) ? state_w[idx] : 0.0f;
}

// ---------------------------------------------------------------------------
// Kernel A: out_s GEMM via V_WMMA_F32_16X16X4_F32 + energy materialization.
// Grid: 256 blocks x 256 threads. Each block: 8 waves x 16 rows = 128 (b,t) rows.
// ---------------------------------------------------------------------------
__global__ __launch_bounds__(256) void crf_gemm_energy(
    const float* __restrict__ inp,      // (B*T, 768)
    const float* __restrict__ mask,     // (B*T)
    const float* __restrict__ wpad,     // (32, 768) zero-padded weights
    const float* __restrict__ state_b,  // (31)
    const float* __restrict__ trans,    // (31, 31)
    float* __restrict__ energy_out,     // (B*T, 961)
    float* __restrict__ outs_ws)        // (B*T, 32)  scratch
{
    __shared__ float sTrans[NL * NL];       // 3.8 KB
    __shared__ float sOuts[128][LPAD];      // 16 KB

    const int tid  = threadIdx.x;
    const int wave = tid >> 5;
    const int lane = tid & 31;

    for (int i = tid; i < NL * NL; i += 256) sTrans[i] = trans[i];

    const int rowBase = (blockIdx.x * 8 + wave) * 16;   // first global row of tile
    const int m     = lane & 15;          // row-within-tile for A, col-within-ntile for B
    const int koff  = (lane >> 4) * 2;    // 0 or 2 (A/B frag K sub-offset)
    const int rhalf = (lane >> 4) * 8;    // accumulator row offset

    const float* aRow = inp  + (size_t)(rowBase + m) * D_DIM + koff;

    const int c0 = m;            // N-tile 0: cols 0..15
    const int c1 = 16 + m;       // N-tile 1: cols 16..31 (31 = zero pad row)
    const float* b0p = wpad + (size_t)c0 * D_DIM + koff;
    const float* b1p = wpad + (size_t)c1 * D_DIM + koff;

    const float sb0 = state_b[c0];
    const float sb1 = (c1 < NL) ? state_b[c1] : 0.0f;

    v8f acc0 = {};
    v8f acc1 = {};

    for (int k = 0; k < D_DIM; k += 4) {
        // A frag: input is streamed once -> non-temporal
        v2f a  = __builtin_nontemporal_load((const v2f*)(aRow + k));
        // B frags: tiny, heavily reused -> regular cached loads (L0/L2 resident)
        v2f b0 = *(const v2f*)(b0p + k);
        v2f b1 = *(const v2f*)(b1p + k);

        acc0 = __builtin_amdgcn_wmma_f32_16x16x4_f32(
                   false, a, false, b0, (short)0, acc0, false, false);
        acc1 = __builtin_amdgcn_wmma_f32_16x16x4_f32(
                   false, a, false, b1, (short)0, acc1, false, false);
    }

    // Spill accumulators (with bias + EOS mask fixup) to LDS + out_s scratch.
    const int blkRow = blockIdx.x * 128;
#pragma unroll
    for (int v = 0; v < 8; ++v) {
        const int rloc = wave * 16 + v + rhalf;   // 0..127 within block
        const int gr   = blkRow + rloc;           // global (b,t) row

        float o0 = acc0[v] + sb0;
        float o1 = acc1[v] + sb1;
        if (c1 == IDX_EOS) {
            const float mk = mask[gr];
            o1 += (mk == 0.0f) ? 20000.0f : 0.0f;
        }
        sOuts[rloc][c0] = o0;
        sOuts[rloc][c1] = o1;
        outs_ws[(size_t)gr * LPAD + c0] = o0;
        outs_ws[(size_t)gr * LPAD + c1] = o1;
    }
    __syncthreads();

    // Energy materialization: 128 rows x 961 floats, streaming NT writes.
    for (int r = 0; r < 128; ++r) {
        float* erow = energy_out + (size_t)(blkRow + r) * (NL * NL);
        for (int e = tid; e < NL * NL; e += 256) {
            const int j = e % NL;
            __builtin_nontemporal_store(sTrans[e] + sOuts[r][j], erow + e);
        }
    }
}

// ---------------------------------------------------------------------------
// Kernel B: forward-algorithm scan + tgt_energy + loss. 64 blocks x 1 wave32.
// Lane j owns label j; trans[:,j] kept in registers.
// ---------------------------------------------------------------------------
__global__ __launch_bounds__(32) void crf_scan(
    const float* __restrict__ outs_ws,  // (B*T, 32)
    const float* __restrict__ trans,    // (31, 31)
    const int*   __restrict__ target,   // (B, T)
    float* __restrict__ loss)           // (B)
{
    __shared__ float sT[NL][LPAD];
    __shared__ float sPart[LPAD];

    const int b = blockIdx.x;
    const int j = threadIdx.x;

    // trans column j into registers; full trans into LDS for the target gather.
    float tc[NL];
#pragma unroll
    for (int i = 0; i < NL; ++i) {
        const float tv = (j < NL) ? trans[i * NL + j] : NEGV;
        tc[i] = tv;
        sT[i][j] = tv;
    }
    __syncthreads();

    const float* orow = outs_ws + (size_t)b * T_DIM * LPAD;

    // part0[j] = trans[BOS, j] + out_s[b, 0, j]
    float pj = (j < NL) ? (tc[IDX_BOS] + orow[j]) : NINF;

    for (int t = 1; t < T_DIM; ++t) {
        sPart[j] = pj;
        __syncthreads();
        float v[NL];
#pragma unroll
        for (int i = 0; i < NL; ++i) v[i] = tc[i] + sPart[i];
        float mx = NINF;
#pragma unroll
        for (int i = 0; i < NL; ++i) mx = fmaxf(mx, v[i]);
        float s = 0.0f;
#pragma unroll
        for (int i = 0; i < NL; ++i) s += __expf(v[i] - mx);
        pj = orow[(size_t)t * LPAD + j] + mx + __logf(s);
        __syncthreads();
    }

    // tgt_energy: sum_t trans[prev_t, tgt_t] + out_s[b, t, tgt_t], lanes stride t.
    float acc = 0.0f;
    for (int t = j; t < T_DIM; t += 32) {
        const int tgt = target[b * T_DIM + t];
        const int prv = (t == 0) ? IDX_BOS : target[b * T_DIM + t - 1];
        acc += sT[prv][tgt] + orow[(size_t)t * LPAD + tgt];
    }

    // loss[b] = logsumexp_j(trans[j, EOS] + part[j]) - tgt_energy
    const float te  = (j < NL) ? trans[j * NL + IDX_EOS] : NINF;
    float val = (j < NL) ? (te + pj) : NINF;

    float mx = val;
    for (int off = 16; off; off >>= 1) mx = fmaxf(mx, __shfl_xor(mx, off));
    float s = __expf(val - mx);
    for (int off = 16; off; off >>= 1) s += __shfl_xor(s, off);
    for (int off = 16; off; off >>= 1) acc += __shfl_xor(acc, off);

    if (j == 0) loss[b] = mx + __logf(s) - acc;
}

// ---------------------------------------------------------------------------
// Launch. Inputs (setup_inputs order): input, target, mask, state_w, state_b,
// trans. Output: loss (64 floats) ++ energy (64*512*31*31 floats).
// Workspace layout: [wpad: 32*768 f32 = 96 KB][outs: 32768*32 f32 = 4 MB].
// ---------------------------------------------------------------------------
extern "C" void kernel_launch(void* const* d_in, const int* in_sizes, int n_in,
                              void* d_out, int out_size, void* d_ws, size_t ws_size,
                              hipStream_t stream) {
    (void)in_sizes; (void)n_in; (void)out_size; (void)ws_size;

    const float* inp     = (const float*)d_in[0];
    const int*   target  = (const int*)  d_in[1];
    const float* mask    = (const float*)d_in[2];
    const float* state_w = (const float*)d_in[3];
    const float* state_b = (const float*)d_in[4];
    const float* trans   = (const float*)d_in[5];

    float* out     = (float*)d_out;
    float* loss    = out;          // 64 floats
    float* energy  = out + B_DIM;  // (B*T, 961)

    float* wpad    = (float*)d_ws;            // (32, 768)
    float* outs_ws = wpad + LPAD * D_DIM;     // (B*T, 32)

    crf_prep_weights<<<dim3((LPAD * D_DIM) / 256), dim3(256), 0, stream>>>(
        state_w, wpad);

    crf_gemm_energy<<<dim3(256), dim3(256), 0, stream>>>(
        inp, mask, wpad, state_b, trans, energy, outs_ws);

    crf_scan<<<dim3(B_DIM), dim3(32), 0, stream>>>(
        outs_ws, trans, target, loss);
}